// GPTrustTransfer_4896262717521
// MI455X (gfx1250) — compile-verified
//
#include <hip/hip_runtime.h>
#include <hip/hip_bf16.h>

typedef float v2f __attribute__((ext_vector_type(2)));
typedef float v8f __attribute__((ext_vector_type(8)));

#define TT 32
#define NN 4096
#define DD 256
#define RR 2

// ---------------------------------------------------------------------------
// Kernel 1: projection  xs[n][t][r] = sum_d obs[t][n][d] * A[r][d]
//           xp[n][r]    = sum_d pred[n][d]   * A[r][d]
// One wave per 16-row tile of the flattened [T*N + N, D] row space.
// Uses V_WMMA_F32_16X16X4_F32 (exact fp32 matrix math).
// ---------------------------------------------------------------------------
__global__ __launch_bounds__(256) void gp_proj_kernel(
    const float* __restrict__ obs,   // [T, N, D]
    const float* __restrict__ pred,  // [N, D]
    const float* __restrict__ A,     // [R, D]
    float* __restrict__ xs,          // [N, T, R]  (ws)
    float* __restrict__ xp)          // [N, R]     (ws)
{
    const int OBS_TILES = (TT * NN) / 16;               // 8192
    int wave = (blockIdx.x * blockDim.x + threadIdx.x) >> 5;
    int lane = threadIdx.x & 31;
    int half = lane >> 4;                               // 0 or 1
    int l    = lane & 15;

    bool is_pred = (wave >= OBS_TILES);
    long tile    = is_pred ? (wave - OBS_TILES) : wave;
    const float* src = is_pred ? pred : obs;
    long row0 = tile * 16;

    // A-fragment row for this lane (16x4 fp32 layout: VGPR v holds K = 2*half + v)
    const float* arow = src + (row0 + l) * DD;
    // B-fragment: B[k][col] = A[col][d0+k]; only cols 0..R-1 are real.
    bool  bvalid = (l < RR);
    const float* brow = A + (bvalid ? l : 0) * DD;
    float bmask = bvalid ? 1.0f : 0.0f;
    int   kbase = half * 2;

    v8f acc = {};
    #pragma unroll 8
    for (int d0 = 0; d0 < DD; d0 += 4) {
        v2f a = *(const v2f*)(arow + d0 + kbase);       // K = kbase, kbase+1
        v2f bfrag;
        bfrag.x = brow[d0 + kbase]     * bmask;
        bfrag.y = brow[d0 + kbase + 1] * bmask;
        acc = __builtin_amdgcn_wmma_f32_16x16x4_f32(
            /*neg_a=*/false, a, /*neg_b=*/false, bfrag,
            /*c_mod=*/(short)0, acc, /*reuse_a=*/false, /*reuse_b=*/false);
    }

    // C/D layout: VGPR v holds row (8*half + v), column = l.
    if (bvalid) {
        int r = l;
        if (is_pred) {
            #pragma unroll
            for (int v = 0; v < 8; v++) {
                long grow = row0 + half * 8 + v;        // global row = n
                xp[grow * RR + r] = acc[v];
            }
        } else {
            #pragma unroll
            for (int v = 0; v < 8; v++) {
                long grow = row0 + half * 8 + v;        // global row = t*N + n
                int t = (int)(grow >> 12);              // N = 4096 = 2^12
                int n = (int)(grow & (NN - 1));
                xs[((long)n * TT + t) * RR + r] = acc[v];
            }
        }
    }
}

// ---------------------------------------------------------------------------
// Kernel 2: sequential sparse-online-GP recurrence. One wave32 per task.
// Lane i owns basis point i, alpha[i], and covariance row C[i][0..31] in regs.
// All broadcasts use wave-uniform lane indices -> v_readlane_b32 to SGPR
// (no LDS / ds_bpermute traffic); only reductions use lane-varying shuffles.
// ---------------------------------------------------------------------------
__device__ __forceinline__ float readlane_f(float v, int srclane) {
    return __int_as_float(__builtin_amdgcn_readlane(__float_as_int(v), srclane));
}

__device__ __forceinline__ float wave_sum(float v) {
    v += __shfl_xor(v, 16);
    v += __shfl_xor(v, 8);
    v += __shfl_xor(v, 4);
    v += __shfl_xor(v, 2);
    v += __shfl_xor(v, 1);
    return v;
}

__global__ __launch_bounds__(256) void gp_recur_kernel(
    const float* __restrict__ xs,    // [N, T, R]
    const float* __restrict__ xp,    // [N, R]
    const int*   __restrict__ perf,  // [T, N, 1]
    const float* __restrict__ by_p,
    const float* __restrict__ nv_p,
    const float* __restrict__ phi_p, // [R]
    const float* __restrict__ s_p,
    float* __restrict__ out)         // [N, 1]
{
    const float SQRT1_2 = 0.70710678118654752f;
    const float CONSTL  = 0.79788456080286536f;   // 2/sqrt(2*pi)

    int wave = (blockIdx.x * blockDim.x + threadIdx.x) >> 5;
    int lane = threadIdx.x & 31;
    if (wave >= NN) return;
    int n = wave;

    // per-lane observation point (basis candidate) and label
    float x0 = xs[((long)n * TT + lane) * RR + 0];
    float x1 = xs[((long)n * TT + lane) * RR + 1];
    float y  = (perf[(long)lane * NN + n] == 1) ? -1.0f : 1.0f;
    float xq0 = xp[n * RR + 0];
    float xq1 = xp[n * RR + 1];

    float mx    = by_p[0];
    float noise = __expf(nv_p[0]) + 0.01f;
    float il0   = __expf(-phi_p[0]);
    float il1   = __expf(-phi_p[1]);
    float sv_   = s_p[0];
    float kstar = sv_ * sv_;

    float alpha = 0.0f;
    float C[TT];
    #pragma unroll
    for (int j = 0; j < TT; j++) C[j] = 0.0f;

    for (int t = 0; t < TT; t++) {
        // broadcast step-t observation/label: uniform index -> v_readlane
        float xt0 = readlane_f(x0, t);
        float xt1 = readlane_f(x1, t);
        float yt  = readlane_f(y, t);

        float d0 = (x0 - xt0) * il0;
        float d1 = (x1 - xt1) * il1;
        float k  = (lane < t) ? (kstar * __expf(-(d0 * d0 + d1 * d1))) : 0.0f;

        float m = wave_sum(k * alpha) - mx;

        // Ck_i = sum_j C[i][j] * k[j];  k[j] broadcast via readlane (SGPR fmac)
        float ck = 0.0f;
        #pragma unroll
        for (int j = 0; j < TT; j++)
            ck = fmaf(C[j], readlane_f(k, j), ck);

        float kck = wave_sum(k * ck);
        float s2  = fmaxf(kstar + kck + noise, 1e-6f);
        float sx  = __builtin_sqrtf(s2);
        float z   = yt * m / sx;
        float Erfz  = 0.5f * (erff(z * SQRT1_2) + 1.0f);
        float ratio = __expf(-0.5f * z * z) * CONSTL / Erfz;
        float q = fminf(fmaxf((yt / sx) * ratio, -1.0f), 1.0f);
        float r = (-z * ratio - ratio * ratio) / s2;
        if (t == 0) {
            q = (yt - mx) / kstar;
            r = -1.0f / (kstar + noise);
        }

        float svec = ck + ((lane == t) ? 1.0f : 0.0f);
        alpha = fmaf(q, svec, alpha);
        float rsv = r * svec;
        #pragma unroll
        for (int j = 0; j < TT; j++)
            C[j] = fmaf(rsv, readlane_f(svec, j), C[j]);
    }

    // GPpredict at xq
    float d0 = (x0 - xq0) * il0;
    float d1 = (x1 - xq1) * il1;
    float kq = kstar * __expf(-(d0 * d0 + d1 * d1));

    float msum = wave_sum(kq * alpha) + mx;
    float ckq = 0.0f;
    #pragma unroll
    for (int j = 0; j < TT; j++)
        ckq = fmaf(C[j], readlane_f(kq, j), ckq);
    float quad = wave_sum(kq * ckq);
    float s2 = kstar + quad + noise;
    float p = 0.5f * (erff((msum / __builtin_sqrtf(s2)) * SQRT1_2) + 1.0f);

    // zero-task fallback: first projected observation all-zero
    float z00 = readlane_f(x0, 0);
    float z01 = readlane_f(x1, 0);
    if (z00 == 0.0f && z01 == 0.0f)
        p = 0.5f * (erff((mx / __builtin_sqrtf(kstar + noise)) * SQRT1_2) + 1.0f);

    p = fminf(fmaxf(p, 0.01f), 0.99f);
    if (lane == 0) out[n] = p;
}

// ---------------------------------------------------------------------------
extern "C" void kernel_launch(void* const* d_in, const int* in_sizes, int n_in,
                              void* d_out, int out_size, void* d_ws, size_t ws_size,
                              hipStream_t stream) {
    const float* obs   = (const float*)d_in[0];  // [T,N,D]
    const float* pred  = (const float*)d_in[1];  // [N,D]
    const float* A     = (const float*)d_in[2];  // [R,D]
    const float* by_p  = (const float*)d_in[3];
    const float* nv_p  = (const float*)d_in[4];
    const float* phi_p = (const float*)d_in[5];  // [R]
    const float* s_p   = (const float*)d_in[6];
    const int*   perf  = (const int*)d_in[7];    // [T,N,1]
    float* out = (float*)d_out;

    float* xs = (float*)d_ws;                    // N*T*R floats
    float* xp = xs + (long)NN * TT * RR;         // N*R floats

    // Projection: (T*N + N)/16 tiles, one wave each, 8 waves per block.
    int total_waves = (TT * NN + NN) / 16;       // 8448
    int blocks1 = (total_waves * 32 + 255) / 256;
    gp_proj_kernel<<<blocks1, 256, 0, stream>>>(obs, pred, A, xs, xp);

    // Recurrence: one wave per task.
    int blocks2 = (NN * 32 + 255) / 256;
    gp_recur_kernel<<<blocks2, 256, 0, stream>>>(xs, xp, perf, by_p, nv_p,
                                                 phi_p, s_p, out);
}